// _RGCNConvWithEdgeType_82781199663113
// MI455X (gfx1250) — compile-verified
//
#include <hip/hip_runtime.h>
#include <hip/hip_bf16.h>

// ---------------------------------------------------------------------------
// RGCNConv (1 relation, basis decomposition)
//   W   = sum_b comp[b] * bases[b]                      [128,128]
//   agg = scatter_add(x[src] -> dst)                    [N,128]
//   out = agg @ W + x @ root + bias                     [N,128]  (f32)
// N = 50000 (= 16*3125), E = 640000, D = 128, N_BASES = 8
//
// Cost model (MI455X): x and agg are each 25.6 MB -> both L2-resident
// (192 MB L2). The 82M f32 scatter atomics dominate; GEMM is 3.3 GFLOP on
// the exact-precision V_WMMA_F32_16X16X4_F32 path.
// ---------------------------------------------------------------------------

typedef __attribute__((ext_vector_type(2))) float v2f;
typedef __attribute__((ext_vector_type(4))) float v4f;
typedef __attribute__((ext_vector_type(8))) float v8f;

#define D 128
#define NBASES 8

// --- Kernel 1: W[i] = sum_b comp[b] * bases[b][i], i in [0, 128*128) -------
__global__ __launch_bounds__(256) void rgcn_weight_kernel(
    const float* __restrict__ bases,   // [8,128,128]
    const float* __restrict__ comp,    // [1,8]
    float* __restrict__ W)             // [128,128]
{
    int i = blockIdx.x * 256 + threadIdx.x;   // grid sized exactly to 16384
    float acc = 0.0f;
#pragma unroll
    for (int b = 0; b < NBASES; ++b)
        acc = fmaf(comp[b], bases[b * (D * D) + i], acc);
    W[i] = acc;
}

// --- Kernel 2: agg[dst[e], :] += x[src[e], :] ------------------------------
// One wave32 per edge; block = (32,8) -> 8 edges per block.
// threadIdx.y is wave-uniform (blockDim.x == wave32 width), so the edge id
// can be pinned to an SGPR with readfirstlane -> indices come in via scalar
// s_load_b64 instead of redundant per-lane vector loads by all 32 lanes of the edge.
// Each lane covers 4 contiguous channels: 1x global_load_b128 + 4 atomics.
__global__ __launch_bounds__(256) void rgcn_scatter_kernel(
    const float* __restrict__ x,          // [N,128]
    const long long* __restrict__ src,    // [E]
    const long long* __restrict__ dst,    // [E]
    float* __restrict__ agg)              // [N,128]
{
    const int lane = threadIdx.x;                       // 0..31
    int e = blockIdx.x * 8 + threadIdx.y;               // edge id (wave-uniform)
    e = __builtin_amdgcn_readfirstlane(e);              // force SGPR -> s_load

    const int s = (int)src[e];
    const int d = (int)dst[e];

    const int c = lane * 4;                             // channel base
    const v4f v = *(const v4f*)(x + (long long)s * D + c);
    float* p = agg + (long long)d * D + c;
    atomicAdd(p + 0, v.x);
    atomicAdd(p + 1, v.y);
    atomicAdd(p + 2, v.z);
    atomicAdd(p + 3, v.w);
}

// --- Kernel 3: out = agg@W + x@root + bias via V_WMMA_F32_16X16X4_F32 ------
// One wave32 per 16x16 output tile. Block = (32,8): 8 N-tiles (covers all
// 128 output columns); blockIdx.x = M-tile (3125 tiles, no tail -> EXEC all
// ones as WMMA requires).
//
// A fragment (16x4, f32): lane%16 = M row; lanes 0-15 hold K={k,k+1},
//   lanes 16-31 hold K={k+2,k+3} in VGPRs {0,1} (a.x, a.y).
// B fragment (4x16, f32): lane%16 = N col; same K split as A.
// C/D (16x16 f32, 8 VGPRs): VGPR i -> row (i + 8*(lane/16)), col lane%16.
__global__ __launch_bounds__(256) void rgcn_gemm_kernel(
    const float* __restrict__ agg,   // [N,128]
    const float* __restrict__ x,     // [N,128]
    const float* __restrict__ W,     // [128,128]
    const float* __restrict__ root,  // [128,128]
    const float* __restrict__ bias,  // [128]
    float* __restrict__ out)         // [N,128]
{
    const int lane = threadIdx.x;          // 0..31
    const int n0   = threadIdx.y * 16;     // this wave's N-tile
    const int m0   = blockIdx.x * 16;      // this block's M-tile

    const int half = lane >> 4;            // 0: K pair {k,k+1}; 1: {k+2,k+3}
    const int l16  = lane & 15;
    const long long rowA = (long long)(m0 + l16) * D;  // A row base (agg / x)
    const int colB = n0 + l16;                         // B column

    v8f c = {};   // f32 accumulator tile

#pragma unroll 4
    for (int k = 0; k < D; k += 4) {
        const int ka = k + 2 * half;
        v2f a, b;

        // agg @ W
        a.x = agg[rowA + ka];
        a.y = agg[rowA + ka + 1];
        b.x = W[ka * D + colB];
        b.y = W[(ka + 1) * D + colB];
        c = __builtin_amdgcn_wmma_f32_16x16x4_f32(
                /*neg_a=*/false, a, /*neg_b=*/false, b,
                /*c_mod=*/(short)0, c, /*reuse_a=*/false, /*reuse_b=*/false);

        // x @ root
        a.x = x[rowA + ka];
        a.y = x[rowA + ka + 1];
        b.x = root[ka * D + colB];
        b.y = root[(ka + 1) * D + colB];
        c = __builtin_amdgcn_wmma_f32_16x16x4_f32(
                false, a, false, b, (short)0, c, false, false);
    }

    // Epilogue: add bias, store D tile per the documented C/D VGPR layout.
    const float bv = bias[colB];
#pragma unroll
    for (int i = 0; i < 8; ++i) {
        const int row = m0 + half * 8 + i;
        out[(long long)row * D + colB] = c[i] + bv;
    }
}

// ---------------------------------------------------------------------------
extern "C" void kernel_launch(void* const* d_in, const int* in_sizes, int n_in,
                              void* d_out, int out_size, void* d_ws, size_t ws_size,
                              hipStream_t stream)
{
    const float*     x          = (const float*)d_in[0];       // [N,128] f32
    const long long* edge_index = (const long long*)d_in[1];   // [2,E] i64
    const float*     bases      = (const float*)d_in[2];       // [8,128,128]
    const float*     comp       = (const float*)d_in[3];       // [1,8]
    const float*     root       = (const float*)d_in[4];       // [128,128]
    const float*     bias       = (const float*)d_in[5];       // [128]
    float*           out        = (float*)d_out;               // [N,128]

    const int       N = in_sizes[0] / D;          // 50000
    const long long E = (long long)in_sizes[1] / 2;

    const long long* src = edge_index;            // edge_index[0]
    const long long* dst = edge_index + E;        // edge_index[1]

    // Workspace layout: W [16384 f32] | agg [N*128 f32]  (~25.7 MB total)
    float* W   = (float*)d_ws;
    float* agg = W + (D * D);

    // agg must be zero at the start of every call (graph-capturable memset).
    hipMemsetAsync(agg, 0, (size_t)N * D * sizeof(float), stream);

    // 1) Combine bases into W.
    rgcn_weight_kernel<<<(D * D) / 256, 256, 0, stream>>>(bases, comp, W);

    // 2) Scatter-add messages: one wave per edge, 4 channels per lane.
    dim3 sblock(32, 8);
    rgcn_scatter_kernel<<<(int)(E / 8), sblock, 0, stream>>>(x, src, dst, agg);

    // 3) Fused dual-GEMM epilogue on the WMMA pipe (exact f32 math).
    dim3 gblock(32, 8);                 // 8 wave32s = 8 N-tiles
    rgcn_gemm_kernel<<<N / 16, gblock, 0, stream>>>(agg, x, W, root, bias, out);
}